// PLIF_62869731279024
// MI455X (gfx1250) — compile-verified
//
#include <hip/hip_runtime.h>

// PLIF (parametric leaky-integrate-and-fire) forward scan over T.
//   mem = w*mem + x_t ; spike = (mem >= 1) ; mem = spike ? 0 : mem
// Pure HBM-streaming workload: 256 MB of traffic, ~11 us at 23.3 TB/s.
// Strategy: one float4 per thread, full unroll over T=16 so all 16
// global_load_b128's are independent and in flight, non-temporal cache
// hints on both loads and stores (stream > 192 MB L2, zero reuse).

typedef float f32x4 __attribute__((ext_vector_type(4)));

namespace {
constexpr int B = 16, T = 16, C = 128, H = 32, W = 32;
constexpr int CHW     = C * H * W;   // 131072 floats: stride between timesteps
constexpr int CHW4    = CHW / 4;     // 32768 float4 = 2^15
constexpr int THREADS = 256;         // 8 wave32 per block
constexpr int TOTAL4  = B * CHW4;    // 524288 float4-sites
}

__global__ __launch_bounds__(THREADS)
void plif_scan_kernel(const float* __restrict__ xp,
                      const float* __restrict__ wp,
                      float*       __restrict__ op) {
  const int gid = blockIdx.x * THREADS + threadIdx.x;   // [0, B*CHW4)
  const int b   = gid >> 15;                            // gid / CHW4
  const int r   = gid & (CHW4 - 1);                     // gid % CHW4

  const f32x4* __restrict__ x4 = (const f32x4*)xp;
  f32x4*       __restrict__ o4 = (f32x4*)op;

  // Single 64-bit base per thread; all 16 t-offsets (t*512KB <= 7.86MB)
  // fit in the signed 24-bit VMEM immediate offset field.
  const size_t base = (size_t)b * (size_t)(T * CHW4) + (size_t)r;

  const float w0 = wp[0];              // scalar leak (s_load, wave-uniform)
  f32x4 mem = {0.0f, 0.0f, 0.0f, 0.0f};

#pragma unroll
  for (int t = 0; t < T; ++t) {
    // Streamed once, never re-read -> non-temporal (th:TH_LOAD_NT).
    f32x4 xt = __builtin_nontemporal_load(x4 + base + (size_t)t * CHW4);

    f32x4 sp;
#pragma unroll
    for (int i = 0; i < 4; ++i) {
      float m  = fmaf(w0, mem[i], xt[i]);   // v_fmac_f32
      bool  s  = (m >= 1.0f);               // v_cmp_ge_f32 (>= v_th)
      sp[i]    = s ? 1.0f : 0.0f;           // v_cndmask
      mem[i]   = s ? 0.0f : m;              // hard reset, v_cndmask
    }

    // Output also written once -> non-temporal store (th:TH_STORE_NT).
    __builtin_nontemporal_store(sp, o4 + base + (size_t)t * CHW4);
  }
}

extern "C" void kernel_launch(void* const* d_in, const int* in_sizes, int n_in,
                              void* d_out, int out_size, void* d_ws, size_t ws_size,
                              hipStream_t stream) {
  (void)in_sizes; (void)n_in; (void)out_size; (void)d_ws; (void)ws_size;
  const float* x = (const float*)d_in[0];   // [B,T,C,H,W] f32
  const float* w = (const float*)d_in[1];   // scalar leak, 1 element
  float* out = (float*)d_out;               // [B,T,C,H,W] f32 spikes

  dim3 grid(TOTAL4 / THREADS);              // 2048 blocks
  dim3 block(THREADS);                      // 256 threads = 8 waves
  plif_scan_kernel<<<grid, block, 0, stream>>>(x, w, out);
}